// IndustrialFNOBlock_51762945852005
// MI455X (gfx1250) — compile-verified
//
#include <hip/hip_runtime.h>
#include <hip/hip_bf16.h>
#include <math.h>

// ---------------------------------------------------------------------------
// IndustrialFNOBlock on MI455X (gfx1250), fp32 WMMA 16x16x4 everywhere.
// Pipeline: gen DFT basis -> G1 fwd partial DFT (WMMA) -> spectral mix (VALU,
// bandwidth bound on 67MB weights) -> G3 inv partial DFT (WMMA) -> PW conv
// (WMMA) -> assemble + depthwise + t-bias + RMSNorm -> fused SwiGLU (WMMA)
// with residual + transpose back to (B,C,L).
// GEMM tiles are staged into LDS by the Tensor Data Mover (double-buffered),
// synchronized with s_wait_tensorcnt + workgroup barriers.
// ---------------------------------------------------------------------------

typedef __attribute__((ext_vector_type(2))) float v2f;
typedef __attribute__((ext_vector_type(8))) float v8f;
typedef __attribute__((ext_vector_type(4))) unsigned int v4u;
typedef __attribute__((ext_vector_type(4))) int v4i;
typedef __attribute__((ext_vector_type(8))) int v8i;

#define B_ 8
#define C_ 128
#define L_ 8192
#define MODES_ 512
#define TDIM_ 256
#define HID_ 352
#define TWO_PI 6.28318530717958647692f

#if __has_builtin(__builtin_amdgcn_tensor_load_to_lds)
#define HAVE_TDM 1
#else
#define HAVE_TDM 0
#endif

__device__ __forceinline__ v8f wmma4(v2f a, v2f b, v8f c) {
  // D = A(16x4,f32) * B(4x16,f32) + C(16x16,f32)
  return __builtin_amdgcn_wmma_f32_16x16x4_f32(false, a, false, b, (short)0, c,
                                               false, false);
}

#if HAVE_TDM
// Workgroup-relative LDS byte offset of a generic pointer to __shared__.
__device__ __forceinline__ unsigned lds_offset(const void* p) {
  return (unsigned)(unsigned long long)(
      const __attribute__((address_space(3))) void*)p;
}

// TDM: load a 2D tile of f32 (tileX elems per row, tileY rows, row stride
// strideX elems in global) into LDS at ldsOff, padding the LDS destination
// by (padAmt+1) DWORDs after every 2^padInt*2 DWORDs (descriptor encoding).
// Descriptor layout per CDNA5 ISA sec. 8 (D# groups 0/1; groups 2/3 zero for
// 2D tiles).  Tensor dims == tile dims -> no OOB handling engaged.
__device__ __forceinline__ void tdm_load_2d(unsigned ldsOff, const float* gsrc,
                                            unsigned tileX, unsigned tileY,
                                            unsigned strideX, unsigned padInt,
                                            unsigned padAmt) {
  const unsigned long long ga = (unsigned long long)gsrc;
  v4u g0;
  g0[0] = 1u;  // count=1, user descriptor, gather off
  g0[1] = ldsOff;
  g0[2] = (unsigned)(ga & 0xffffffffu);
  g0[3] = (unsigned)((ga >> 32) & 0x01ffffffu) | (2u << 30);  // type=2
  v8i g1;
  g1[0] = (int)((2u << 16)            // data_size = 4B
                | (1u << 20)          // pad_enable
                | (padInt << 22) | (padAmt << 25));
  g1[1] = (int)((tileX & 0xffffu) << 16);            // tensor_dim0 lo16
  g1[2] = (int)((tileX >> 16) | ((tileY & 0xffffu) << 16));  // dim0 hi/dim1 lo
  g1[3] = (int)((tileY >> 16) | (tileX << 16));      // dim1 hi | tile_dim0
  g1[4] = (int)tileY;                                 // tile_dim1, tile_dim2=0
  g1[5] = (int)strideX;                               // tensor_dim0_stride lo32
  g1[6] = 0;                                          // stride0 hi, stride1 lo
  g1[7] = 0;
  const v4i z4 = {0, 0, 0, 0};
#if defined(__clang_major__) && __clang_major__ >= 23
  const v8i z8 = {0, 0, 0, 0, 0, 0, 0, 0};
  __builtin_amdgcn_tensor_load_to_lds(g0, g1, z4, z4, z8, 0);
#else
  __builtin_amdgcn_tensor_load_to_lds(g0, g1, z4, z4, 0);
#endif
}
#endif  // HAVE_TDM

// ---------------------------------------------------------------------------
// Basis generator: Bf[n][m<512] = cos(2pi n m / L), Bf[n][512+m] = -sin(...)
// Binv = Bf^T (row-major 1024 x 8192). Scales are folded into Dcomb later.
// ---------------------------------------------------------------------------
__global__ void gen_basis(float* __restrict__ Bf, float* __restrict__ Binv) {
  long id = (long)blockIdx.x * blockDim.x + threadIdx.x;
  if (id >= (long)L_ * 1024) return;
  int n = (int)(id >> 10);
  int col = (int)(id & 1023);
  int m = col & 511;
  int phase = (m * n) & (L_ - 1);
  float ang = (float)phase * (TWO_PI / (float)L_);
  float s, c;
  __sincosf(ang, &s, &c);
  float v = (col < 512) ? c : -s;
  Bf[(long)n * 1024 + col] = v;
  Binv[(long)col * L_ + n] = v;
}

// ---------------------------------------------------------------------------
// Weight transposes so WMMA B-operands are coalesced (K-major rows).
// ---------------------------------------------------------------------------
__global__ void transpose_w(const float* __restrict__ w1,
                            const float* __restrict__ w2,
                            const float* __restrict__ w3,
                            float* __restrict__ w1T, float* __restrict__ w2T,
                            float* __restrict__ w3T) {
  int id = blockIdx.x * blockDim.x + threadIdx.x;
  if (id >= HID_ * C_) return;
  int j = id / C_, c = id % C_;
  w1T[c * HID_ + j] = w1[id];
  w2T[c * HID_ + j] = w2[id];
  int o = id / HID_, jj = id % HID_;
  w3T[jj * C_ + o] = w3[id];
}

// ---------------------------------------------------------------------------
// t_bias[b][c] = sum_k silu(t_emb[b,k]) tw[c,k] + tb[c]
// ---------------------------------------------------------------------------
__global__ __launch_bounds__(128) void time_bias(
    const float* __restrict__ t_emb, const float* __restrict__ tw,
    const float* __restrict__ tb, float* __restrict__ tbias) {
  __shared__ float se[TDIM_];
  const int b = blockIdx.x, t = threadIdx.x;
  for (int k = t; k < TDIM_; k += 128) {
    float v = t_emb[b * TDIM_ + k];
    se[k] = v / (1.0f + __expf(-v));
  }
  __syncthreads();
  float acc = 0.f;
  for (int k = 0; k < TDIM_; ++k) acc += se[k] * tw[t * TDIM_ + k];
  tbias[b * C_ + t] = acc + tb[t];
}

// ---------------------------------------------------------------------------
// Generic fp32 WMMA GEMM: C(MxN) = A(MxK) @ B(KxN), all row-major,
// M,N multiples of 64, K multiple of 16.  Block = 256 thr (8 waves),
// 64x64 C-tile, wave -> 32x16 (two 16x16 accumulators).
// A/B tiles staged into padded LDS by TDM (double-buffered, wave0 issues,
// s_wait_tensorcnt 2 overlaps next-tile DMA with current-tile WMMA).
// ---------------------------------------------------------------------------
__global__ __launch_bounds__(256) void wmma_gemm_f32(
    const float* __restrict__ A, const float* __restrict__ B,
    float* __restrict__ C, int M, int N, int K, long aStride, long bStride,
    long cStride) {
  __shared__ __attribute__((aligned(16))) float As[2][64][20];
  __shared__ __attribute__((aligned(16))) float Bs[2][16][68];
  (void)M;
  const int bz = blockIdx.z;
  A += (long)bz * aStride;
  B += (long)bz * bStride;
  C += (long)bz * cStride;
  const int nBase = blockIdx.x * 64;
  const int mBase = blockIdx.y * 64;
  const int t = threadIdx.x;
  const int wave = t >> 5, lane = t & 31;
  const int half = lane >> 4, l15 = lane & 15;
  const int mt = (wave >> 2) * 32;  // 0 or 32
  const int nt = (wave & 3) * 16;   // 0..48
  v8f acc0 = {0.f, 0.f, 0.f, 0.f, 0.f, 0.f, 0.f, 0.f};
  v8f acc1 = {0.f, 0.f, 0.f, 0.f, 0.f, 0.f, 0.f, 0.f};

#if HAVE_TDM
  const unsigned ldsA0 = lds_offset(&As[0][0][0]);
  const unsigned ldsA1 = lds_offset(&As[1][0][0]);
  const unsigned ldsB0 = lds_offset(&Bs[0][0][0]);
  const unsigned ldsB1 = lds_offset(&Bs[1][0][0]);
  if (t < 32) {  // wave 0 drives the Tensor Data Mover
    tdm_load_2d(ldsA0, A + (long)mBase * K, 16u, 64u, (unsigned)K, 3u, 3u);
    tdm_load_2d(ldsB0, B + nBase, 64u, 16u, (unsigned)N, 5u, 3u);
  }
  int buf = 0;
  for (int k0 = 0; k0 < K; k0 += 16) {
    __syncthreads();  // everyone done reading buf^1 from previous iteration
    if (t < 32) {
      if (k0 + 16 < K) {
        const unsigned la = buf ? ldsA0 : ldsA1;
        const unsigned lb = buf ? ldsB0 : ldsB1;
        tdm_load_2d(la, A + (long)mBase * K + (k0 + 16), 16u, 64u, (unsigned)K,
                    3u, 3u);
        tdm_load_2d(lb, B + (long)(k0 + 16) * N + nBase, 64u, 16u, (unsigned)N,
                    5u, 3u);
        __builtin_amdgcn_s_wait_tensorcnt(2);  // current pair done, next flies
      } else {
        __builtin_amdgcn_s_wait_tensorcnt(0);
      }
    }
    __syncthreads();
#pragma unroll
    for (int kk = 0; kk < 16; kk += 4) {
      v2f bf;
      bf[0] = Bs[buf][kk + 2 * half][nt + l15];
      bf[1] = Bs[buf][kk + 2 * half + 1][nt + l15];
      v2f a0;
      a0[0] = As[buf][mt + l15][kk + 2 * half];
      a0[1] = As[buf][mt + l15][kk + 2 * half + 1];
      v2f a1;
      a1[0] = As[buf][mt + 16 + l15][kk + 2 * half];
      a1[1] = As[buf][mt + 16 + l15][kk + 2 * half + 1];
      acc0 = wmma4(a0, bf, acc0);
      acc1 = wmma4(a1, bf, acc1);
    }
    buf ^= 1;
  }
#else
  const int aRow = t >> 2, aCol = (t & 3) * 4;   // stage A 64x16
  const int bRow = t >> 4, bCol = (t & 15) * 4;  // stage B 16x64
  for (int k0 = 0; k0 < K; k0 += 16) {
    const float4 av =
        *(const float4*)(A + (long)(mBase + aRow) * K + k0 + aCol);
    const float4 bv =
        *(const float4*)(B + (long)(k0 + bRow) * N + nBase + bCol);
    if (k0 + 16 < K) {
      __builtin_prefetch(A + (long)(mBase + aRow) * K + k0 + 16 + aCol, 0, 3);
      __builtin_prefetch(B + (long)(k0 + 16 + bRow) * N + nBase + bCol, 0, 3);
    }
    __syncthreads();
    *(float4*)&As[0][aRow][aCol] = av;
    *(float4*)&Bs[0][bRow][bCol] = bv;
    __syncthreads();
#pragma unroll
    for (int kk = 0; kk < 16; kk += 4) {
      v2f bf;
      bf[0] = Bs[0][kk + 2 * half][nt + l15];
      bf[1] = Bs[0][kk + 2 * half + 1][nt + l15];
      v2f a0;
      a0[0] = As[0][mt + l15][kk + 2 * half];
      a0[1] = As[0][mt + l15][kk + 2 * half + 1];
      v2f a1;
      a1[0] = As[0][mt + 16 + l15][kk + 2 * half];
      a1[1] = As[0][mt + 16 + l15][kk + 2 * half + 1];
      acc0 = wmma4(a0, bf, acc0);
      acc1 = wmma4(a1, bf, acc1);
    }
  }
#endif
#pragma unroll
  for (int v = 0; v < 8; ++v) {
    const int row = mBase + mt + v + 8 * half;
    const int col = nBase + nt + l15;
    C[(long)row * N + col] = acc0[v];
    C[(long)(row + 16) * N + col] = acc1[v];
  }
}

// ---------------------------------------------------------------------------
// Spectral mix: Y[b,o,m] = sum_c X[b,c,m] * W[c,o,m]  (complex), rotate by
// exp(-i 2pi shifts[o] m), subtract X[b,o,m], fold irfft scale, write Dcomb.
// Bandwidth bound on 67MB of W; 16 consecutive modes -> coalesced reads.
// ---------------------------------------------------------------------------
__global__ __launch_bounds__(256) void spectral_mix(
    const float* __restrict__ XF, const float* __restrict__ wr,
    const float* __restrict__ wi, const float* __restrict__ shifts,
    float* __restrict__ D) {
  __shared__ float xr[B_][16];
  __shared__ float xi[B_][16];
  const int m0 = blockIdx.x * 16;
  const int o0 = blockIdx.y * 16;
  const int t = threadIdx.x;
  const int ml = t & 15, ol = t >> 4;
  const int m = m0 + ml, o = o0 + ol;
  float accR[B_], accI[B_];
#pragma unroll
  for (int b = 0; b < B_; ++b) {
    accR[b] = 0.f;
    accI[b] = 0.f;
  }
  const int lb = t >> 5;
  const int lw = (t >> 4) & 1;
  const int lm = t & 15;
  for (int c = 0; c < C_; ++c) {
    __syncthreads();
    {
      const long row = ((long)lb * C_ + c) * 1024;
      if (lw == 0)
        xr[lb][lm] = XF[row + m0 + lm];
      else
        xi[lb][lm] = XF[row + 512 + m0 + lm];
    }
    __syncthreads();
    const long widx = ((long)c * C_ + o) * MODES_ + m;
    const float wR = wr[widx];
    const float wI = wi[widx];
    if (c + 1 < C_) {
      __builtin_prefetch(wr + widx + (long)C_ * MODES_, 0, 3);
      __builtin_prefetch(wi + widx + (long)C_ * MODES_, 0, 3);
    }
#pragma unroll
    for (int b = 0; b < B_; ++b) {
      const float xR = xr[b][ml], xI = xi[b][ml];
      accR[b] += xR * wR - xI * wI;
      accI[b] += xR * wI + xI * wR;
    }
  }
  float sr, cr;
  __sincosf(-TWO_PI * shifts[o] * (float)m, &sr, &cr);
  const float scale = ((m == 0) ? 1.0f : 2.0f) / (float)L_;
#pragma unroll
  for (int b = 0; b < B_; ++b) {
    const long row = ((long)b * C_ + o) * 1024;
    const float yR = accR[b] * cr - accI[b] * sr;
    const float yI = accR[b] * sr + accI[b] * cr;
    D[row + m] = (yR - XF[row + m]) * scale;
    D[row + 512 + m] = (yI - XF[row + 512 + m]) * scale;
  }
}

// ---------------------------------------------------------------------------
// Assemble h_pre = Yg + PW + pw_b + depthwise(x) + dw_b + t_bias, transpose
// a (32 l x 128 c) tile in LDS, RMSNorm over c, write hN in (B,L,C) layout.
// ---------------------------------------------------------------------------
__global__ __launch_bounds__(256) void assemble_norm(
    const float* __restrict__ Yg, const float* __restrict__ PW,
    const float* __restrict__ x, const float* __restrict__ dw_w,
    const float* __restrict__ dw_b, const float* __restrict__ pw_b,
    const float* __restrict__ tbias, const float* __restrict__ norm_w,
    float* __restrict__ hN) {
  __shared__ float T[32][132];
  __shared__ float P[32][8];
  __shared__ float S[32];
  const int b = blockIdx.y;
  const int l0 = blockIdx.x * 32;
  const int t = threadIdx.x;
  {
    const int c = t >> 1;
    const int seg = (t & 1) * 16;
    const long base = ((long)b * C_ + c) * L_ + l0 + seg;
    const float w0 = dw_w[c * 3 + 0], w1 = dw_w[c * 3 + 1],
                w2 = dw_w[c * 3 + 2];
    const float addc = dw_b[c] + pw_b[c] + tbias[b * C_ + c];
    for (int i = 0; i < 16; ++i) {
      const long l = l0 + seg + i;
      const float xm = (l > 0) ? x[base + i - 1] : 0.f;
      const float xc = x[base + i];
      const float xp = (l < L_ - 1) ? x[base + i + 1] : 0.f;
      T[seg + i][c] =
          Yg[base + i] + PW[base + i] + w0 * xm + w1 * xc + w2 * xp + addc;
    }
  }
  __syncthreads();
  {
    const int row = t >> 3, part = t & 7;
    float s = 0.f;
    for (int i = 0; i < 16; ++i) {
      const float v = T[row][part * 16 + i];
      s += v * v;
    }
    P[row][part] = s;
  }
  __syncthreads();
  if (t < 32) {
    float s = 0.f;
    for (int i = 0; i < 8; ++i) s += P[t][i];
    S[t] = rsqrtf(s * (1.0f / (float)C_) + 1e-6f);
  }
  __syncthreads();
  {
    const int row = t >> 3, part = t & 7;
    const float sc = S[row];
    const long ob = ((long)b * L_ + l0 + row) * C_;
    for (int i = 0; i < 16; ++i) {
      const int c = part * 16 + i;
      hN[ob + c] = T[row][c] * sc * norm_w[c];
    }
  }
}

// ---------------------------------------------------------------------------
// Fused SwiGLU + residual + transpose back to (B,C,L).
// Block: 64 L-rows.  Phase1: U1=h@w1T, U2=h@w2T per 16x16 tile (WMMA),
// V=silu(U1)*U2 in LDS.  Phase2: Out=V@w3T (WMMA).  Store transposed + x.
// hN tile staged by TDM when available.
// ---------------------------------------------------------------------------
__global__ __launch_bounds__(256) void swiglu_out(
    const float* __restrict__ hN, const float* __restrict__ w1T,
    const float* __restrict__ w2T, const float* __restrict__ w3T,
    const float* __restrict__ x, float* __restrict__ out) {
  __shared__ __attribute__((aligned(16))) float As[64][132];  // h tile / Vout
  __shared__ float V[64][356];
  const int b = blockIdx.y;
  const int l0 = blockIdx.x * 64;
  const int t = threadIdx.x;
  const int wave = t >> 5, lane = t & 31;
  const int half = lane >> 4, l15 = lane & 15;

#if HAVE_TDM
  if (t < 32) {
    tdm_load_2d(lds_offset(&As[0][0]), hN + ((long)(b * L_ + l0) << 7), 128u,
                64u, 128u, 6u, 3u);
    __builtin_amdgcn_s_wait_tensorcnt(0);
  }
  __syncthreads();
#else
  for (int idx = t * 4; idx < 64 * C_; idx += 256 * 4) {
    const int row = idx >> 7, col = idx & 127;
    const float4 v =
        *(const float4*)(hN + ((long)(b * L_ + l0 + row) << 7) + col);
    *(float4*)&As[row][col] = v;
  }
  __syncthreads();
#endif

  // Phase 1: 4 m-tiles x 22 n-tiles = 88 tiles over 8 waves.
  for (int tt = wave; tt < 88; tt += 8) {
    const int mt = (tt & 3) * 16;
    const int nt = (tt >> 2) * 16;
    v8f u1 = {0.f, 0.f, 0.f, 0.f, 0.f, 0.f, 0.f, 0.f};
    v8f u2 = {0.f, 0.f, 0.f, 0.f, 0.f, 0.f, 0.f, 0.f};
    for (int k = 0; k < C_; k += 4) {
      v2f a;
      a[0] = As[mt + l15][k + 2 * half];
      a[1] = As[mt + l15][k + 2 * half + 1];
      v2f b1;
      b1[0] = w1T[(k + 2 * half) * HID_ + nt + l15];
      b1[1] = w1T[(k + 2 * half + 1) * HID_ + nt + l15];
      v2f b2;
      b2[0] = w2T[(k + 2 * half) * HID_ + nt + l15];
      b2[1] = w2T[(k + 2 * half + 1) * HID_ + nt + l15];
      u1 = wmma4(a, b1, u1);
      u2 = wmma4(a, b2, u2);
    }
#pragma unroll
    for (int v = 0; v < 8; ++v) {
      float g = u1[v];
      g = g / (1.0f + __expf(-g));
      V[mt + v + 8 * half][nt + l15] = g * u2[v];
    }
  }
  __syncthreads();

  // Phase 2: 4 m-tiles x 8 n-tiles = 32 tiles over 8 waves.
  for (int tt = wave; tt < 32; tt += 8) {
    const int mt = (tt & 3) * 16;
    const int nt = (tt >> 2) * 16;
    v8f acc = {0.f, 0.f, 0.f, 0.f, 0.f, 0.f, 0.f, 0.f};
    for (int k = 0; k < HID_; k += 4) {
      v2f a;
      a[0] = V[mt + l15][k + 2 * half];
      a[1] = V[mt + l15][k + 2 * half + 1];
      v2f bf;
      bf[0] = w3T[(k + 2 * half) * C_ + nt + l15];
      bf[1] = w3T[(k + 2 * half + 1) * C_ + nt + l15];
      acc = wmma4(a, bf, acc);
    }
#pragma unroll
    for (int v = 0; v < 8; ++v) As[mt + v + 8 * half][nt + l15] = acc[v];
  }
  __syncthreads();

  // Transposed store with residual: out[b][c][l] = Vout[l][c] + x[b][c][l]
  {
    const int c = t >> 1, h2 = t & 1;
    const long gb = ((long)b * C_ + c) * L_ + l0 + h2 * 32;
    for (int i = 0; i < 32; i += 4) {
      const float4 xv = *(const float4*)(x + gb + i);
      float4 r;
      r.x = As[h2 * 32 + i + 0][c] + xv.x;
      r.y = As[h2 * 32 + i + 1][c] + xv.y;
      r.z = As[h2 * 32 + i + 2][c] + xv.z;
      r.w = As[h2 * 32 + i + 3][c] + xv.w;
      *(float4*)(out + gb + i) = r;
    }
  }
}

// ---------------------------------------------------------------------------
extern "C" void kernel_launch(void* const* d_in, const int* in_sizes, int n_in,
                              void* d_out, int out_size, void* d_ws,
                              size_t ws_size, hipStream_t stream) {
  (void)in_sizes;
  (void)n_in;
  (void)out_size;
  (void)ws_size;
  const float* x = (const float*)d_in[0];
  const float* t_emb = (const float*)d_in[1];
  const float* w_real = (const float*)d_in[2];
  const float* w_imag = (const float*)d_in[3];
  const float* shifts = (const float*)d_in[4];
  const float* pw_w = (const float*)d_in[5];
  const float* pw_b = (const float*)d_in[6];
  const float* dw_w = (const float*)d_in[7];
  const float* dw_b = (const float*)d_in[8];
  const float* norm_w = (const float*)d_in[9];
  const float* w1 = (const float*)d_in[10];
  const float* w2 = (const float*)d_in[11];
  const float* w3 = (const float*)d_in[12];
  const float* tw = (const float*)d_in[13];
  const float* tb = (const float*)d_in[14];

  float* ws = (float*)d_ws;
  float* Bf = ws;                // 8192*1024
  float* Binv = Bf + 8388608L;   // 1024*8192
  float* XF = Binv + 8388608L;   // 1024*1024
  float* Dcomb = XF + 1048576L;  // 1024*1024
  float* Yg = Dcomb + 1048576L;  // 1024*8192
  float* PWb = Yg + 8388608L;    // 1024*8192
  float* hN = PWb + 8388608L;    // 8*8192*128
  float* tbias = hN + 8388608L;  // 1024
  float* w1T = tbias + 1024L;    // 128*352
  float* w2T = w1T + 45056L;
  float* w3T = w2T + 45056L;

  gen_basis<<<32768, 256, 0, stream>>>(Bf, Binv);
  transpose_w<<<(HID_ * C_ + 255) / 256, 256, 0, stream>>>(w1, w2, w3, w1T,
                                                           w2T, w3T);
  time_bias<<<B_, 128, 0, stream>>>(t_emb, tw, tb, tbias);

  // G1: XF(1024x1024) = x(1024x8192) @ Bf(8192x1024)
  wmma_gemm_f32<<<dim3(16, 16, 1), 256, 0, stream>>>(x, Bf, XF, 1024, 1024,
                                                     8192, 0, 0, 0);
  // PW: per-batch (128x8192) = pw_w(128x128) @ x_b(128x8192)
  wmma_gemm_f32<<<dim3(128, 2, 8), 256, 0, stream>>>(
      pw_w, x, PWb, 128, 8192, 128, 0, (long)C_ * L_, (long)C_ * L_);

  spectral_mix<<<dim3(32, 8), 256, 0, stream>>>(XF, w_real, w_imag, shifts,
                                                Dcomb);

  // G3: Yg(1024x8192) = Dcomb(1024x1024) @ Binv(1024x8192)
  wmma_gemm_f32<<<dim3(128, 16, 1), 256, 0, stream>>>(Dcomb, Binv, Yg, 1024,
                                                      8192, 1024, 0, 0, 0);

  assemble_norm<<<dim3(256, 8), 256, 0, stream>>>(Yg, PWb, x, dw_w, dw_b, pw_b,
                                                  tbias, norm_w, hN);

  swiglu_out<<<dim3(128, 8), 256, 0, stream>>>(hN, w1T, w2T, w3T, x,
                                               (float*)d_out);
}